// PCFG_67602785239512
// MI455X (gfx1250) — compile-verified
//
#include <hip/hip_runtime.h>
#include <cstddef>

// ---------------------------------------------------------------------------
// PCFG inside algorithm (compound-PCFG style) for MI455X / gfx1250.
//
// All GEMMs (MLP trunk 60x256x256, vocab head 60x10000x256, rule head
// 30x8100x256, per-width inside exp-GEMM (<=1800 x 2700 x 96)) run on
// V_WMMA_F32_16X16X4_F32 (fp32 in / fp32 accum): exact reference numerics,
// and the workload is latency-bound (29 serial CKY widths; ~30 MB of
// intermediates, L2-resident in the 192 MB L2), not compute-bound.
//
// GEMM codegen strategy:
//  * index CLAMPING instead of per-lane load guards -> unconditional loads,
//    no exec-mask branch storms (garbage padding rows/cols of C are never
//    stored; GEMM is row/col separable).
//  * A tile staged to LDS with b128 coalesced loads; fragments read as
//    conflict-free ds_load_b64 (padded stride K+4).
//  * exp(ruleT - ym) built once, stored COLUMN-major so the inside-GEMM B
//    fragment is one aligned global_load_b64 per K-step.
//
// Input flattening assumption: jax pytree order = dicts by sorted key:
//   0 input_ids(240,i32) 1 length(8,i32) 2 nt_emb 3 root_emb
//   root_mlp: 4 lin_in.b 5 lin_in.w 6 lin_out.b 7 lin_out.w
//             res1: 8 u1.b 9 u1.w 10 u2.b 11 u2.w 12 v1.b 13 v1.w
//                   14 v2.b 15 v2.w 16 w.b 17 w.w          res2: 18..27
//   28 rule_mlp.b 29 rule_mlp.w 30 t_emb
//   vocab_mlp: 31 lin_in.b 32 lin_in.w 33 lin_out.b 34 lin_out.w
//              res1: 35..44   res2: 45..54
// ---------------------------------------------------------------------------

typedef float v2f __attribute__((ext_vector_type(2)));
typedef float v4f __attribute__((ext_vector_type(4)));
typedef float v8f __attribute__((ext_vector_type(8)));

#define NEG_INF (-__builtin_inff())

static constexpr int kNT = 30;     // nonterminals
static constexpr int kT  = 60;     // preterminals
static constexpr int kS  = 90;     // NT + T
static constexpr int kV  = 10000;  // vocab
static constexpr int kD  = 256;    // hidden
static constexpr int kB  = 8;      // batch
static constexpr int kN  = 30;     // sentence length
static constexpr int kSS = kNT * kS;     // 2700 columns of ruleT
static constexpr int kSP = 2704;         // 2700 padded to 16
static constexpr int kKP = 96;           // 90 padded to mult of 4 (K of exp-GEMM)
static constexpr int kMaxRows = 1800;    // max spans*(w-1)*B = 225*8

__device__ __forceinline__ int imin(int a, int b) { return a < b ? a : b; }

// ------------------------- generic fused linear (WMMA) ---------------------
// Y = [resid +] [relu]( X(MxK) @ W(KxN) + bias ) ; K % 4 == 0 (template).
// block = (32,4): 4 waves share one 16-row A tile (LDS), each owns a 16-wide
// N sub-tile.
template <int K>
__global__ void linear_wmma_f32(const float* __restrict__ X,
                                const float* __restrict__ W,
                                const float* __restrict__ bias,
                                const float* __restrict__ resid,
                                float* __restrict__ Y,
                                int M, int N, int relu) {
  constexpr int LDK = K + 4;                 // LDS stride: 16B-aligned, no bank conflicts
  __shared__ float As[16 * LDK];
  const int lane = threadIdx.x;
  const int half = lane >> 4, lm = lane & 15;
  const int m0 = blockIdx.x * 16;
  const int n0 = (blockIdx.y * 4 + threadIdx.y) * 16;
  const int tid = threadIdx.y * 32 + threadIdx.x;

  // stage A tile (clamped rows), coalesced b128
  for (int idx = tid; idx < 16 * (K / 4); idx += 128) {
    const int r = idx / (K / 4);
    const int c4 = (idx - r * (K / 4)) * 4;
    const float* src = X + (size_t)imin(m0 + r, M - 1) * K + c4;
    *reinterpret_cast<v4f*>(&As[r * LDK + c4]) =
        *reinterpret_cast<const v4f*>(src);
  }
  __syncthreads();

  const int col = n0 + lm;
  const int colc = imin(col, N - 1);         // clamped: loads always in-bounds
  const float* Bcol = W + colc;
  const float* Afrag = &As[lm * LDK];

  v8f acc;
  for (int t = 0; t < 8; ++t) acc[t] = 0.f;
#pragma unroll 8
  for (int k0 = 0; k0 < K; k0 += 4) {
    const int ka = k0 + 2 * half;            // A: lanes 0-15 K={k0,k0+1}, 16-31 K={k0+2,k0+3}
    const v2f av = *reinterpret_cast<const v2f*>(Afrag + ka);
    v2f bv;
    bv.x = Bcol[(size_t)ka * N];
    bv.y = Bcol[(size_t)(ka + 1) * N];
    acc = __builtin_amdgcn_wmma_f32_16x16x4_f32(false, av, false, bv,
                                                (short)0, acc, false, false);
  }
  if (col < N) {
    const float bval = bias ? bias[col] : 0.f;
    for (int jj = 0; jj < 8; ++jj) {
      const int r = m0 + jj + 8 * half;      // C layout: VGPR j -> M=j (+8 for hi lanes)
      if (r < M) {
        float v = acc[jj] + bval;
        if (relu) v = fmaxf(v, 0.f);
        if (resid) v += resid[(size_t)r * N + col];
        Y[(size_t)r * N + col] = v;
      }
    }
  }
}

// ------------------------- root MLP (M = 1, plain VALU) ---------------------
struct RootPtrs {
  const float* emb;
  const float* li_b; const float* li_w;
  const float* lo_b; const float* lo_w;
  // [0]=u1.b [1]=u1.w [2]=u2.b [3]=u2.w [4]=v1.b [5]=v1.w [6]=v2.b [7]=v2.w [8]=w.b [9]=w.w
  const float* r1[10];
  const float* r2[10];
};

__device__ __forceinline__ void lin256(const float* __restrict__ xin,
                                       float* __restrict__ xout,
                                       const float* __restrict__ w,
                                       const float* __restrict__ b,
                                       bool relu, const float* __restrict__ resid) {
  const int j = threadIdx.x;
  float s = b[j];
  for (int k = 0; k < kD; ++k) s += xin[k] * w[k * kD + j];
  if (relu) s = fmaxf(s, 0.f);
  if (resid) s += resid[j];
  xout[j] = s;
  __syncthreads();
}

__global__ void root_mlp_kernel(RootPtrs rp, float* __restrict__ roots) {
  __shared__ float A[kD], Bf[kD], C[kD];
  __shared__ float red[2];
  const int j = threadIdx.x;
  A[j] = rp.emb[j];
  __syncthreads();
  lin256(A, Bf, rp.li_w, rp.li_b, false, nullptr);        // h
  // res1
  lin256(Bf, A, rp.r1[9], rp.r1[8], false, nullptr);      // a
  lin256(A, C, rp.r1[1], rp.r1[0], true, nullptr);        // relu(a@u1)
  lin256(C, Bf, rp.r1[5], rp.r1[4], true, A);             // x1 = a + relu(.@v1)
  lin256(Bf, C, rp.r1[3], rp.r1[2], true, nullptr);       // relu(x1@u2)
  lin256(C, A, rp.r1[7], rp.r1[6], true, Bf);             // x2 = x1 + relu(.@v2)
  // res2
  lin256(A, Bf, rp.r2[9], rp.r2[8], false, nullptr);
  lin256(Bf, C, rp.r2[1], rp.r2[0], true, nullptr);
  lin256(C, A, rp.r2[5], rp.r2[4], true, Bf);
  lin256(A, C, rp.r2[3], rp.r2[2], true, nullptr);
  lin256(C, Bf, rp.r2[7], rp.r2[6], true, A);             // final hidden in Bf
  // lin_out (256 -> 30) + log_softmax
  if (j < kNT) {
    float o = rp.lo_b[j];
    for (int k = 0; k < kD; ++k) o += Bf[k] * rp.lo_w[k * kNT + j];
    C[j] = o;
  }
  __syncthreads();
  if (j == 0) {
    float m = NEG_INF;
    for (int nt = 0; nt < kNT; ++nt) m = fmaxf(m, C[nt]);
    float ss = 0.f;
    for (int nt = 0; nt < kNT; ++nt) ss += expf(C[nt] - m);
    red[0] = m; red[1] = logf(ss);
  }
  __syncthreads();
  if (j < kNT) roots[j] = C[j] - red[0] - red[1];
}

// ------------------------- row reductions ----------------------------------
__global__ void row_lse_kernel(const float* __restrict__ in, float* __restrict__ lse, int N) {
  __shared__ float sh[256];
  const int row = blockIdx.x, tid = threadIdx.x;
  const float* p = in + (size_t)row * N;
  float m = NEG_INF;
  for (int c = tid; c < N; c += 256) m = fmaxf(m, p[c]);
  sh[tid] = m; __syncthreads();
  for (int s = 128; s > 0; s >>= 1) { if (tid < s) sh[tid] = fmaxf(sh[tid], sh[tid + s]); __syncthreads(); }
  m = sh[0]; __syncthreads();
  float ss = 0.f;
  for (int c = tid; c < N; c += 256) ss += expf(p[c] - m);
  sh[tid] = ss; __syncthreads();
  for (int s = 128; s > 0; s >>= 1) { if (tid < s) sh[tid] += sh[tid + s]; __syncthreads(); }
  if (tid == 0) lse[row] = m + logf(sh[0]);
}

__global__ void row_logsoftmax_kernel(const float* __restrict__ in, float* __restrict__ out, int N) {
  __shared__ float sh[256];
  __shared__ float stat[2];
  const int row = blockIdx.x, tid = threadIdx.x;
  const float* p = in + (size_t)row * N;
  float m = NEG_INF;
  for (int c = tid; c < N; c += 256) m = fmaxf(m, p[c]);
  sh[tid] = m; __syncthreads();
  for (int s = 128; s > 0; s >>= 1) { if (tid < s) sh[tid] = fmaxf(sh[tid], sh[tid + s]); __syncthreads(); }
  m = sh[0]; __syncthreads();
  float ss = 0.f;
  for (int c = tid; c < N; c += 256) ss += expf(p[c] - m);
  sh[tid] = ss; __syncthreads();
  for (int s = 128; s > 0; s >>= 1) { if (tid < s) sh[tid] += sh[tid + s]; __syncthreads(); }
  if (tid == 0) { stat[0] = m; stat[1] = logf(sh[0]); }
  __syncthreads();
  const float sub = stat[0] + stat[1];
  for (int c = tid; c < N; c += 256) out[(size_t)row * N + c] = p[c] - sub;
}

// exp(ruleT - ym): loop-invariant operand of every inside GEMM.
// Stored COLUMN-major: expRuleT[col * kKP + s1] -> B fragment is one b64 load.
__global__ void build_ruleT_kernel(const float* __restrict__ rulesLog,
                                   float* __restrict__ expRuleT,
                                   float* __restrict__ ymPad) {
  const int col = blockIdx.x * blockDim.x + threadIdx.x;   // col = nt*90 + s2
  if (col >= kSS) return;
  const int nt = col / kS, s2 = col - nt * kS;
  const float* base = rulesLog + (size_t)nt * kS * kS + s2;
  float ym = NEG_INF;
  for (int s1 = 0; s1 < kS; ++s1) ym = fmaxf(ym, base[s1 * kS]);
  ymPad[col] = ym;
  for (int s1 = 0; s1 < kKP; ++s1)
    expRuleT[(size_t)col * kKP + s1] = (s1 < kS) ? expf(base[s1 * kS] - ym) : 0.f;
}

// chart init: children[i][i][b][NT+t] = emission_logit[t][ids[b][i]] - lse[t]
__global__ void init_children_kernel(const int* __restrict__ ids,
                                     const float* __restrict__ vlog,
                                     const float* __restrict__ vlse,
                                     float* __restrict__ ch) {
  const int idx = blockIdx.x * 256 + threadIdx.x;
  if (idx >= kN * kN * kB * kS) return;
  const int s = idx % kS;
  int r = idx / kS;
  const int b = r & 7; r >>= 3;
  const int i2 = r % kN, i1 = r / kN;
  float v = NEG_INF;
  if (i1 == i2 && s >= kNT) {
    const int t = s - kNT;
    const int id = ids[b * kN + i1];
    v = vlog[(size_t)t * kV + id] - vlse[t];
  }
  ch[idx] = v;
}

// ------------------------- inside loop kernels ------------------------------
// row index = ((i*(w-1) + (u-1))*8 + b)
__global__ void inside_prep_kernel(const float* __restrict__ children,
                                   float* __restrict__ xExp,
                                   float* __restrict__ xm, int w) {
  __shared__ float sh[128];
  const int rowI = blockIdx.x;
  const int b = rowI & 7;
  const int iu = rowI >> 3;
  const int u = iu % (w - 1) + 1;
  const int i = iu / (w - 1);
  const float* left = children + (((size_t)i * kN + (i + u - 1)) * kB + b) * kS;
  const int tid = threadIdx.x;
  const float v = (tid < kS) ? left[tid] : NEG_INF;
  sh[tid] = v; __syncthreads();
  for (int s = 64; s > 0; s >>= 1) { if (tid < s) sh[tid] = fmaxf(sh[tid], sh[tid + s]); __syncthreads(); }
  float m = sh[0];
  if (!(m > NEG_INF && m < __builtin_inff())) m = 0.f;     // reference isfinite guard
  if (tid < kKP) xExp[(size_t)rowI * kKP + tid] = (tid < kS) ? expf(v - m) : 0.f;
  if (tid == 0) xm[rowI] = m;
}

// outLog[r][col] = log(xExp[r] @ expRuleT[:,col]) + xm[r] + ym[col] + right[r][s2]
__global__ void inside_gemm_kernel(const float* __restrict__ xExp,
                                   const float* __restrict__ xm,
                                   const float* __restrict__ expRuleT,
                                   const float* __restrict__ ymPad,
                                   const float* __restrict__ children,
                                   float* __restrict__ outLog,
                                   int rows, int w) {
  constexpr int K = kKP;                     // 96
  constexpr int LDK = K + 4;
  __shared__ float As[16 * LDK];
  const int lane = threadIdx.x;
  const int half = lane >> 4, lm = lane & 15;
  const int m0 = blockIdx.x * 16;
  const int n0 = (blockIdx.y * 4 + threadIdx.y) * 16;
  const int tid = threadIdx.y * 32 + threadIdx.x;

  for (int idx = tid; idx < 16 * (K / 4); idx += 128) {
    const int r = idx / (K / 4);
    const int c4 = (idx - r * (K / 4)) * 4;
    const float* src = xExp + (size_t)imin(m0 + r, rows - 1) * K + c4;
    *reinterpret_cast<v4f*>(&As[r * LDK + c4]) =
        *reinterpret_cast<const v4f*>(src);
  }
  __syncthreads();

  const int col = n0 + lm;
  const int colc = imin(col, kSP - 1);
  const float* Bcol = expRuleT + (size_t)colc * K;   // column-major
  const float* Afrag = &As[lm * LDK];

  v8f acc;
  for (int t = 0; t < 8; ++t) acc[t] = 0.f;
#pragma unroll 8
  for (int k0 = 0; k0 < K; k0 += 4) {
    const int ka = k0 + 2 * half;
    const v2f av = *reinterpret_cast<const v2f*>(Afrag + ka);
    const v2f bv = *reinterpret_cast<const v2f*>(Bcol + ka);
    acc = __builtin_amdgcn_wmma_f32_16x16x4_f32(false, av, false, bv,
                                                (short)0, acc, false, false);
  }
  if (col < kSS) {
    const int s2 = col % kS;
    const float ym = ymPad[col];
    for (int jj = 0; jj < 8; ++jj) {
      const int r = m0 + jj + 8 * half;
      if (r < rows) {
        const int b = r & 7;
        const int iu = r >> 3;
        const int u = iu % (w - 1) + 1;
        const int i = iu / (w - 1);
        const float right =
            children[(((size_t)(i + u) * kN + (i + w - 1)) * kB + b) * kS + s2];
        outLog[(size_t)r * kSP + col] = logf(acc[jj]) + xm[r] + ym + right;
      }
    }
  }
}

// score[i][b][nt] = logsumexp over (u, s2) of outLog; write into chart[:NT].
__global__ void inside_reduce_kernel(const float* __restrict__ outLog,
                                     float* __restrict__ children, int w) {
  __shared__ float sh[128];
  const int idx = blockIdx.x;
  const int nt = idx % kNT;
  const int b = (idx / kNT) & 7;
  const int i = idx / (kNT * kB);
  const int tid = threadIdx.x;
  const int count = (w - 1) * kS;
  float m = NEG_INF;
  for (int t = tid; t < count; t += 128) {
    const int u1 = t / kS, s2 = t - u1 * kS;
    const int r = (i * (w - 1) + u1) * kB + b;
    m = fmaxf(m, outLog[(size_t)r * kSP + nt * kS + s2]);
  }
  sh[tid] = m; __syncthreads();
  for (int s = 64; s > 0; s >>= 1) { if (tid < s) sh[tid] = fmaxf(sh[tid], sh[tid + s]); __syncthreads(); }
  m = sh[0]; __syncthreads();
  float score;
  if (m > NEG_INF) {
    float ss = 0.f;
    for (int t = tid; t < count; t += 128) {
      const int u1 = t / kS, s2 = t - u1 * kS;
      const int r = (i * (w - 1) + u1) * kB + b;
      ss += expf(outLog[(size_t)r * kSP + nt * kS + s2] - m);
    }
    sh[tid] = ss; __syncthreads();
    for (int s = 64; s > 0; s >>= 1) { if (tid < s) sh[tid] += sh[tid + s]; __syncthreads(); }
    score = m + logf(sh[0]);
  } else {
    score = NEG_INF;
  }
  if (tid == 0)
    children[(((size_t)i * kN + (i + w - 1)) * kB + b) * kS + nt] = score;
}

__global__ void final_kernel(const float* __restrict__ ch,
                             const float* __restrict__ roots,
                             const int* __restrict__ length,
                             float* __restrict__ out) {
  const int b = threadIdx.x;
  if (b >= kB) return;
  const int L = length[b];
  const float* f = ch + (((size_t)0 * kN + (L - 1)) * kB + b) * kS;
  float vals[kNT];
  float m = NEG_INF;
  for (int nt = 0; nt < kNT; ++nt) { vals[nt] = roots[nt] + f[nt]; m = fmaxf(m, vals[nt]); }
  float o = NEG_INF;
  if (m > NEG_INF) {
    float ss = 0.f;
    for (int nt = 0; nt < kNT; ++nt) ss += expf(vals[nt] - m);
    o = m + logf(ss);
  }
  out[b] = o;
}

// ------------------------------ host ---------------------------------------
extern "C" void kernel_launch(void* const* d_in, const int* in_sizes, int n_in,
                              void* d_out, int out_size, void* d_ws, size_t ws_size,
                              hipStream_t stream) {
  (void)in_sizes; (void)n_in; (void)out_size; (void)ws_size;
  auto F = [&](int i) { return (const float*)d_in[i]; };
  const int* ids    = (const int*)d_in[0];
  const int* length = (const int*)d_in[1];

  float* wsf = (float*)d_ws;
  size_t off = 0;
  auto alloc = [&](size_t nfl) { float* p = wsf + off; off += nfl; return p; };
  float* children = alloc((size_t)kN * kN * kB * kS);      // 648000
  float* expRuleT = alloc((size_t)kSP * kKP);              // 259584 (col-major)
  float* ymPad    = alloc(kSP);
  float* xExp     = alloc((size_t)kMaxRows * kKP);
  float* xm       = alloc(kMaxRows + 24);
  float* outLog   = alloc((size_t)kMaxRows * kSP);         // 4.87M floats
  float* vlog     = alloc((size_t)kT * kV);                // 600000
  float* vlse     = alloc(64);
  float* rlog     = alloc((size_t)kNT * kS * kS);          // 243000
  float* rlsm     = alloc((size_t)kNT * kS * kS);
  float* roots    = alloc(32);
  float* buf0 = alloc(kT * kD), *buf1 = alloc(kT * kD), *buf2 = alloc(kT * kD);
  float* buf3 = alloc(kT * kD), *buf4 = alloc(kT * kD);

  auto linear = [&](const float* X, const float* W, const float* b,
                    const float* resid, float* Y, int M, int N, int relu) {
    dim3 blk(32, 4);
    dim3 grd((M + 15) / 16, (N + 63) / 64);
    linear_wmma_f32<kD><<<grd, blk, 0, stream>>>(X, W, b, resid, Y, M, N, relu);
  };

  // ---- roots ----
  RootPtrs rp;
  rp.emb = F(3);
  rp.li_b = F(4); rp.li_w = F(5); rp.lo_b = F(6); rp.lo_w = F(7);
  for (int k = 0; k < 10; ++k) { rp.r1[k] = F(8 + k); rp.r2[k] = F(18 + k); }
  root_mlp_kernel<<<1, kD, 0, stream>>>(rp, roots);

  // ---- vocab MLP trunk (t_emb: 60x256) ----
  linear(F(30), F(32), F(31), nullptr, buf0, kT, kD, 0);   // lin_in
  // res1
  linear(buf0, F(44), F(43), nullptr, buf1, kT, kD, 0);    // a = h @ res1.w
  linear(buf1, F(36), F(35), nullptr, buf2, kT, kD, 1);    // relu(a@u1)
  linear(buf2, F(40), F(39), buf1,    buf3, kT, kD, 1);    // x1 = a + relu(.@v1)
  linear(buf3, F(38), F(37), nullptr, buf2, kT, kD, 1);    // relu(x1@u2)
  linear(buf2, F(42), F(41), buf3,    buf4, kT, kD, 1);    // x2 = x1 + relu(.@v2)
  // res2
  linear(buf4, F(54), F(53), nullptr, buf1, kT, kD, 0);
  linear(buf1, F(46), F(45), nullptr, buf2, kT, kD, 1);
  linear(buf2, F(50), F(49), buf1,    buf3, kT, kD, 1);
  linear(buf3, F(48), F(47), nullptr, buf2, kT, kD, 1);
  linear(buf2, F(52), F(51), buf3,    buf0, kT, kD, 1);
  // vocab head: 60 x 10000
  linear(buf0, F(34), F(33), nullptr, vlog, kT, kV, 0);
  row_lse_kernel<<<kT, 256, 0, stream>>>(vlog, vlse, kV);

  // ---- rules: nt_emb(30x256) @ rule_w(256x8100), log_softmax, exp-ruleT ----
  linear(F(2), F(29), F(28), nullptr, rlog, kNT, kS * kS, 0);
  row_logsoftmax_kernel<<<kNT, 256, 0, stream>>>(rlog, rlsm, kS * kS);
  build_ruleT_kernel<<<(kSS + 127) / 128, 128, 0, stream>>>(rlsm, expRuleT, ymPad);

  // ---- chart init ----
  init_children_kernel<<<(kN * kN * kB * kS + 255) / 256, 256, 0, stream>>>(
      ids, vlog, vlse, children);

  // ---- CKY inside: 29 serial widths ----
  for (int w = 2; w <= kN; ++w) {
    const int spans = kN - w + 1;
    const int rows = spans * (w - 1) * kB;
    inside_prep_kernel<<<rows, 128, 0, stream>>>(children, xExp, xm, w);
    dim3 blk(32, 4);
    dim3 grd((rows + 15) / 16, (kSP + 63) / 64);
    inside_gemm_kernel<<<grd, blk, 0, stream>>>(xExp, xm, expRuleT, ymPad,
                                                children, outLog, rows, w);
    inside_reduce_kernel<<<spans * kB * kNT, 128, 0, stream>>>(outLog, children, w);
  }

  final_kernel<<<1, 32, 0, stream>>>(children, roots, length, (float*)d_out);
}